// Encoder2_37117107372138
// MI455X (gfx1250) — compile-verified
//
#include <hip/hip_runtime.h>
#include <hip/hip_bf16.h>

typedef __attribute__((ext_vector_type(2))) float v2f;
typedef __attribute__((ext_vector_type(8))) float v8f;

#define IN_CH 128
#define LAT   32
#define TILE_M 16
#define APAD  132   // padded LDS row stride (floats) to avoid 16-way bank conflicts

// ---------------- degree / normalization ----------------

__global__ void deg_init_kernel(float* __restrict__ deg, int n) {
    int i = blockIdx.x * blockDim.x + threadIdx.x;
    if (i < n) deg[i] = 1.0f;   // self-loop contributes 1 to every node's degree
}

__global__ void deg_edges_kernel(const int* __restrict__ ei, const int* __restrict__ yei,
                                 float* __restrict__ deg, int e1, int e2) {
    int e = blockIdx.x * blockDim.x + threadIdx.x;
    int et = e1 + e2;
    if (e >= et) return;
    int d = (e < e1) ? ei[e1 + e] : yei[e2 + (e - e1)];   // row 1 = dst
    atomicAdd(deg + d, 1.0f);
}

__global__ void dinv_kernel(float* __restrict__ deg, int n) {
    int i = blockIdx.x * blockDim.x + threadIdx.x;
    if (i < n) deg[i] = rsqrtf(deg[i]);   // deg >= 1 always (self loops)
}

// ---------------- fp32 WMMA GEMM: h = x @ W ----------------
// One wave computes a 16x32 tile of h. K=128 in 32 steps of 4 via
// V_WMMA_F32_16X16X4_F32. 4 waves (128 threads) per block.

__global__ __launch_bounds__(128) void gemm_wmma_kernel(const float* __restrict__ x,
                                                        const float* __restrict__ W,
                                                        float* __restrict__ h, int n) {
    __shared__ float sW[IN_CH * LAT];          // 16 KB, [k*LAT + n]
    __shared__ float sA[4][TILE_M * APAD];     // ~33 KB, per-wave padded x tile

    const int tid  = threadIdx.x;
    const int wave = tid >> 5;
    const int lane = tid & 31;
    const int ntiles = (n + TILE_M - 1) / TILE_M;
    const int tile = blockIdx.x * 4 + wave;

    // Stage W cooperatively (coalesced float4)
    {
        const float4* Wv = (const float4*)W;
        float4* sWv = (float4*)sW;
        for (int i = tid; i < (IN_CH * LAT) / 4; i += 128) sWv[i] = Wv[i];
    }
    // Stage this wave's 16x128 A tile (coalesced float4 per row)
    if (tile < ntiles) {
        const int r0 = tile * TILE_M;
        for (int i = lane; i < TILE_M * (IN_CH / 4); i += 32) {
            int row = i >> 5;            // IN_CH/4 = 32 float4 per row
            int c4  = i & 31;
            int grow = r0 + row; if (grow >= n) grow = n - 1;
            float4 v = ((const float4*)(x + (size_t)grow * IN_CH))[c4];
            float* dst = &sA[wave][row * APAD + c4 * 4];
            dst[0] = v.x; dst[1] = v.y; dst[2] = v.z; dst[3] = v.w;
        }
    }
    __syncthreads();

    if (tile >= ntiles) return;          // wave-uniform: EXEC stays all-1s below

    v8f acc0 = {};                       // cols 0..15
    v8f acc1 = {};                       // cols 16..31
    const int m    = lane & 15;          // A row for this lane
    const int koff = (lane >> 4) << 1;   // lanes 0-15: K+{0,1}; lanes 16-31: K+{2,3}
    const int nn   = lane & 15;          // B/D column within tile

    for (int kk = 0; kk < IN_CH / 4; ++kk) {
        const int kb = kk * 4 + koff;
        v2f a;                           // 32-bit A 16x4 layout (ISA 7.12.2)
        a.x = sA[wave][m * APAD + kb];
        a.y = sA[wave][m * APAD + kb + 1];
        v2f b0, b1;                      // B rows striped across lanes
        b0.x = sW[(kb    ) * LAT + nn];
        b0.y = sW[(kb + 1) * LAT + nn];
        b1.x = sW[(kb    ) * LAT + 16 + nn];
        b1.y = sW[(kb + 1) * LAT + 16 + nn];
        acc0 = __builtin_amdgcn_wmma_f32_16x16x4_f32(false, a, false, b0,
                                                     (short)0, acc0, false, false);
        acc1 = __builtin_amdgcn_wmma_f32_16x16x4_f32(false, a, false, b1,
                                                     (short)0, acc1, false, false);
    }

    // D layout: VGPR r -> M = r (lanes 0-15) / r+8 (lanes 16-31), N = lane&15
    const int r0   = tile * TILE_M;
    const int rofs = (lane >> 4) << 3;
#pragma unroll
    for (int r = 0; r < 8; ++r) {
        int row = r0 + r + rofs;
        if (row < n) {
            h[(size_t)row * LAT + nn]      = acc0[r];
            h[(size_t)row * LAT + 16 + nn] = acc1[r];
        }
    }
}

// ---------------- out init: self-loop message + bias ----------------

__global__ void init_out_kernel(const float* __restrict__ dinv, const float* __restrict__ h,
                                const float* __restrict__ b, float* __restrict__ out, int n) {
    int gid = blockIdx.x * blockDim.x + threadIdx.x;
    if (gid >= n * LAT) return;
    int i = gid >> 5, c = gid & 31;
    float di = dinv[i];
    out[gid] = di * di * h[gid] + b[c];
}

// ---------------- edge scatter: wave32 = one edge, lane = channel ----------------

__global__ void edge_scatter_kernel(const int* __restrict__ ei, const int* __restrict__ yei,
                                    const float* __restrict__ dinv, const float* __restrict__ h,
                                    float* __restrict__ out, int e1, int e2) {
    long long gid = (long long)blockIdx.x * blockDim.x + threadIdx.x;
    int e = (int)(gid >> 5);
    int c = (int)(gid & 31);
    int et = e1 + e2;
    if (e >= et) return;
    int s, d;
    if (e < e1) { s = ei[e];            d = ei[e1 + e]; }
    else        { int f = e - e1; s = yei[f]; d = yei[e2 + f]; }
    float w = dinv[s] * dinv[d];                         // wave-uniform broadcast loads
    atomicAdd(out + (size_t)d * LAT + c, w * h[(size_t)s * LAT + c]);
}

// ---------------- launcher ----------------

extern "C" void kernel_launch(void* const* d_in, const int* in_sizes, int n_in,
                              void* d_out, int out_size, void* d_ws, size_t ws_size,
                              hipStream_t stream) {
    const float* x   = (const float*)d_in[0];
    const int*   ei  = (const int*)  d_in[1];
    const int*   yei = (const int*)  d_in[2];
    const float* W   = (const float*)d_in[3];
    const float* b   = (const float*)d_in[4];
    float* out = (float*)d_out;

    const int n  = in_sizes[0] / IN_CH;
    const int e1 = in_sizes[1] / 2;
    const int e2 = in_sizes[2] / 2;
    const int et = e1 + e2;

    // workspace: deg/dinv [n] floats, then h [n*LAT] floats (256B aligned)
    float* deg = (float*)d_ws;
    size_t hoff = (((size_t)n * sizeof(float)) + 255) & ~(size_t)255;
    float* h = (float*)((char*)d_ws + hoff);

    // 1) degrees (self loop baked in as 1.0)
    deg_init_kernel<<<(n + 255) / 256, 256, 0, stream>>>(deg, n);
    deg_edges_kernel<<<(et + 255) / 256, 256, 0, stream>>>(ei, yei, deg, e1, e2);
    // 2) deg -> deg^{-1/2}
    dinv_kernel<<<(n + 255) / 256, 256, 0, stream>>>(deg, n);
    // 3) h = x @ W via fp32 WMMA (4 tiles of 16 rows per block)
    {
        int ntiles = (n + TILE_M - 1) / TILE_M;
        int blocks = (ntiles + 3) / 4;
        gemm_wmma_kernel<<<blocks, 128, 0, stream>>>(x, W, h, n);
    }
    // 4) out = dinv^2 * h + bias (self-loop term), full overwrite of d_out
    init_out_kernel<<<((size_t)n * LAT + 255) / 256, 256, 0, stream>>>(deg, h, b, out, n);
    // 5) scatter-add normalized edge messages
    {
        long long threads = (long long)et * 32;
        int blocks = (int)((threads + 255) / 256);
        edge_scatter_kernel<<<blocks, 256, 0, stream>>>(ei, yei, deg, h, out, e1, e2);
    }
}